// S4Layer_38792144618291
// MI455X (gfx1250) — compile-verified
//
#include <hip/hip_runtime.h>

// ---------------------------------------------------------------------------
// S4 (DPLR) forward for MI455X (gfx1250, wave32, fp32 WMMA + async-to-LDS).
//   H=512 channels, N=64 states, B=8 batch, L=2048.
// Pipeline:
//   1) gen_tables:    COS/SIN[l][t] = cos/sin(2*pi*(l*t mod L)/L)
//   2) at_roots:      Cauchy-kernel eval at roots of unity -> (H,L) complex
//   3) transpose_u:   u (B,L,H) -> XT[h][b][l]
//   4) idft_wmma:     K = (1/L)(Ar@COS + (-Ai)@SIN), D folded into K[h,0]
//   5) conv_wmma:     Y_h = T_h @ X_h (lower-tri Toeplitz == causal conv;
//                     2L zero-padded FFT conv in the reference is exactly this)
//   6) transpose_out: YT[h][b][t] -> y (B,L,H)
// Matrix math uses V_WMMA_F32_16X16X4_F32 (full fp32 reference precision).
// Hot LDS staging uses GLOBAL_LOAD_ASYNC_TO_LDS_B128 (ISA 15.18.3 op 98,
// ASYNCcnt-tracked) via inline asm + s_wait_asynccnt, so tile fills bypass
// VGPRs entirely; each fill completes before the WMMA compute that reads it.
// Roofline: ~268MB mandatory HBM traffic ~= 11.5us @ 23.3TB/s; WMMA work
// (~5e6 ops) sits below that, so the design centers on coalesced global
// I/O, async LDS fills, and conflict-free LDS strides.
// Workspace: 3*H*L + 2*L*L + 2*B*H*L floats = 113.25 MB of d_ws.
// ---------------------------------------------------------------------------

typedef float v2f __attribute__((ext_vector_type(2)));
typedef float v8f __attribute__((ext_vector_type(8)));

constexpr int Hc = 512;
constexpr int Nc = 64;
constexpr int Bc = 8;
constexpr int Lc = 2048;
constexpr int HL = Hc * Lc;          // 1,048,576
constexpr int LL = Lc * Lc;          // 4,194,304
constexpr int BHL = Bc * Hc * Lc;    // 8,388,608
constexpr float TWO_PI_F = 6.283185307179586f;

// Async DMA: global -> LDS, 16 bytes per active lane, tracked by ASYNCcnt.
// GVS addressing: 64-bit SGPR base + per-lane signed 32-bit byte offset.
// The low 32 bits of a flat LDS pointer are the LDS byte offset (ISA 10.2).
__device__ __forceinline__ void async_b128_to_lds(const float* lds_dst,
                                                  const float* gbase,
                                                  int byte_off) {
  unsigned lds_addr = (unsigned)(unsigned long long)lds_dst;
  asm volatile("global_load_async_to_lds_b128 %0, %1, %2"
               :
               : "v"(lds_addr), "v"(byte_off), "s"(gbase)
               : "memory");
}
__device__ __forceinline__ void async_wait0() {
  asm volatile("s_wait_asynccnt 0" ::: "memory");
}

// ---------------------------------------------------------------- kernel 1 --
__global__ __launch_bounds__(256) void s4_gen_tables(float* __restrict__ C,
                                                     float* __restrict__ S) {
  int idx = blockIdx.x * 256 + threadIdx.x;      // < L*L
  int l = idx >> 11;
  int t = idx & (Lc - 1);
  int ph = (l * t) & (Lc - 1);                   // exact phase mod L
  float s, c;
  __sincosf((float)ph * (TWO_PI_F / (float)Lc), &s, &c);
  C[idx] = c;
  S[idx] = s;
}

// ---------------------------------------------------------------- kernel 2 --
__global__ __launch_bounds__(256) void s4_at_roots(
    const float* __restrict__ Lr, const float* __restrict__ Li,
    const float* __restrict__ Pr, const float* __restrict__ Pi,
    const float* __restrict__ Br, const float* __restrict__ Bi,
    const float* __restrict__ Cr, const float* __restrict__ Ci,
    const float* __restrict__ log_dt,
    float* __restrict__ AR, float* __restrict__ AIN) {
  __shared__ float sLr[Nc], sLi[Nc];
  __shared__ float w00r[Nc], w00i[Nc], w01r[Nc], w01i[Nc];
  __shared__ float w10r[Nc], w10i[Nc], w11r[Nc];
  const int h = blockIdx.x, tid = threadIdx.x;
  if (tid < Nc) {
    const int nn = tid;
    sLr[nn] = Lr[nn];
    sLi[nn] = Li[nn];
    const float cr = Cr[h * Nc + nn], ci = Ci[h * Nc + nn];
    const float br = Br[nn], bi = Bi[nn];
    const float pr = Pr[nn], pi = Pi[nn];
    // conj(C)*B, conj(C)*P, conj(P)*B, conj(P)*P (last is purely real)
    w00r[nn] = cr * br + ci * bi;  w00i[nn] = cr * bi - ci * br;
    w01r[nn] = cr * pr + ci * pi;  w01i[nn] = cr * pi - ci * pr;
    w10r[nn] = pr * br + pi * bi;  w10i[nn] = pr * bi - pi * br;
    w11r[nn] = pr * pr + pi * pi;
  }
  __syncthreads();
  const float gfac = 2.0f * __expf(-log_dt[h]);   // 2/dt
  for (int l = tid; l < Lc; l += 256) {
    float sph, cph;
    __sincosf((float)l * (TWO_PI_F / (float)Lc), &sph, &cph);
    const float omr = cph, omi = -sph;            // omega = exp(-2*pi*i*l/L)
    const float opr = 1.0f + omr, opi = omi;      // 1 + omega
    const float invden = 1.0f / (opr * opr + opi * opi);
    const float c2r = 2.0f * opr * invden;        // c2 = 2/(1+omega)
    const float c2i = -2.0f * opi * invden;
    // g = (2/dt) * (1-omega)/(1+omega)
    const float gr = gfac * (1.0f - (omr * omr + omi * omi)) * invden;
    const float gi = gfac * (-2.0f * omi) * invden;
    float k00r = 0.f, k00i = 0.f, k01r = 0.f, k01i = 0.f;
    float k10r = 0.f, k10i = 0.f, k11r = 0.f, k11i = 0.f;
#pragma unroll 8
    for (int nn = 0; nn < Nc; ++nn) {
      const float dr = gr - sLr[nn], di = gi - sLi[nn];
      const float idn = 1.0f / (dr * dr + di * di);
      const float rr = dr * idn, ri = -di * idn;   // 1/(g - Lambda_n)
      k00r += w00r[nn] * rr - w00i[nn] * ri;  k00i += w00r[nn] * ri + w00i[nn] * rr;
      k01r += w01r[nn] * rr - w01i[nn] * ri;  k01i += w01r[nn] * ri + w01i[nn] * rr;
      k10r += w10r[nn] * rr - w10i[nn] * ri;  k10i += w10r[nn] * ri + w10i[nn] * rr;
      k11r += w11r[nn] * rr;                  k11i += w11r[nn] * ri;
    }
    // at = c2 * (k00 - k01*k10/(1+k11))
    const float e1r = 1.0f + k11r, e1i = k11i;
    const float ide = 1.0f / (e1r * e1r + e1i * e1i);
    const float pnr = k01r * k10r - k01i * k10i;
    const float pni = k01r * k10i + k01i * k10r;
    const float tr = (pnr * e1r + pni * e1i) * ide;
    const float ti = (pni * e1r - pnr * e1i) * ide;
    const float sr = k00r - tr, si = k00i - ti;
    AR[h * Lc + l] = c2r * sr - c2i * si;
    AIN[h * Lc + l] = -(c2r * si + c2i * sr);      // store NEGATED imag part
  }
}

// ---------------------------------------------------------------- kernel 3 --
__global__ __launch_bounds__(256) void s4_transpose_u(const float* __restrict__ u,
                                                      float* __restrict__ XT) {
  __shared__ float tile[64][33];
  const int h0 = blockIdx.x * 32, l0 = blockIdx.y * 64, b = blockIdx.z;
  const int tid = threadIdx.x;
#pragma unroll
  for (int i = 0; i < 8; ++i) {
    int lin = tid + 256 * i;
    int li = lin >> 5, hh = lin & 31;
    tile[li][hh] = u[((size_t)(b * Lc + l0 + li)) * Hc + h0 + hh];
  }
  __syncthreads();
#pragma unroll
  for (int i = 0; i < 8; ++i) {
    int lin = tid + 256 * i;
    int hh = lin >> 6, li = lin & 63;
    XT[((size_t)((h0 + hh) * Bc + b)) * Lc + l0 + li] = tile[li][hh];
  }
}

// ---------------------------------------------------------------- kernel 4 --
// K[h,t] = (1/L) * sum_l (AR[h,l]*cos(2pi*l*t/L) + AIN[h,l]*sin(2pi*l*t/L))
// Tiled fp32 WMMA GEMM: BM=64, BN=64, BK=32, 128 threads (4 waves).
// Tiles land in LDS via GLOBAL_LOAD_ASYNC_TO_LDS_B128 (no VGPR round trip).
__global__ __launch_bounds__(128) void s4_idft_wmma(
    const float* __restrict__ AR, const float* __restrict__ AIN,
    const float* __restrict__ COS, const float* __restrict__ SIN,
    const float* __restrict__ Dd, float* __restrict__ Kout) {
  __shared__ __align__(16) float Ash[64 * 36];   // AR  tile [m][k], stride 36
  __shared__ __align__(16) float Ish[64 * 36];   // AIN tile
  __shared__ __align__(16) float Bcs[32 * 68];   // COS tile [k][n], stride 68
  __shared__ __align__(16) float Bss[32 * 68];   // SIN tile
  const int h0 = blockIdx.x * 64, t0 = blockIdx.y * 64;
  const int tid = threadIdx.x;
  const int w = tid >> 5, lane = tid & 31, n = lane & 15, hi = lane >> 4;
  v8f acc[4] = {};
  for (int kk = 0; kk < Lc; kk += 32) {
#pragma unroll
    for (int i = 0; i < 4; ++i) {                // A tiles: 64x32 = 512 b128/array
      int lin = tid + 128 * i;
      int row = lin >> 3, c4 = (lin & 7) << 2;
      int go = ((h0 + row) * Lc + kk + c4) * 4;  // byte offset
      async_b128_to_lds(&Ash[row * 36 + c4], AR, go);
      async_b128_to_lds(&Ish[row * 36 + c4], AIN, go);
    }
#pragma unroll
    for (int i = 0; i < 4; ++i) {                // B tiles: 32x64 = 512 b128/array
      int lin = tid + 128 * i;
      int row = lin >> 4, c4 = (lin & 15) << 2;
      int go = ((kk + row) * Lc + t0 + c4) * 4;  // byte offset
      async_b128_to_lds(&Bcs[row * 68 + c4], COS, go);
      async_b128_to_lds(&Bss[row * 68 + c4], SIN, go);
    }
    async_wait0();
    __syncthreads();
#pragma unroll
    for (int k = 0; k < 32; k += 4) {
      // ISA 7.12.2: A lane holds M=lane%16; vgpr0/1 = K = (0,1)+2*(lane/16)
      const int kc = k + 2 * hi;
      const int arow = (16 * w + n) * 36;
      v2f a;  a.x = Ash[arow + kc];  a.y = Ash[arow + kc + 1];
      v2f ai; ai.x = Ish[arow + kc]; ai.y = Ish[arow + kc + 1];
#pragma unroll
      for (int nt = 0; nt < 4; ++nt) {
        v2f bc; bc.x = Bcs[kc * 68 + nt * 16 + n]; bc.y = Bcs[(kc + 1) * 68 + nt * 16 + n];
        v2f bs; bs.x = Bss[kc * 68 + nt * 16 + n]; bs.y = Bss[(kc + 1) * 68 + nt * 16 + n];
        acc[nt] = __builtin_amdgcn_wmma_f32_16x16x4_f32(
            false, a, false, bc, (short)0, acc[nt], false, false);
        acc[nt] = __builtin_amdgcn_wmma_f32_16x16x4_f32(
            false, ai, false, bs, (short)0, acc[nt], false, false);
      }
    }
    __syncthreads();
  }
  const float invL = 1.0f / (float)Lc;
#pragma unroll
  for (int nt = 0; nt < 4; ++nt)
#pragma unroll
    for (int r = 0; r < 8; ++r) {                // C/D: row = r + 8*(lane/16)
      const int hh = h0 + 16 * w + 8 * hi + r;
      const int tt = t0 + nt * 16 + n;
      float v = acc[nt][r] * invL;
      if (tt == 0) v += Dd[hh];                  // fold skip connection into K[0]
      Kout[hh * Lc + tt] = v;
    }
}

// ---------------------------------------------------------------- kernel 5 --
// Causal Toeplitz matmul per channel: y[t] = sum_{s<=t} K[t-s] * x[s].
// One block per (h, 64-row t-block); 4 waves, 16 rows each; N-dim = 16
// (8 real batch columns + 8 zero columns). K row and X panels arrive in LDS
// via async DMA; next X panel is prefetched into L2 during compute.
__global__ __launch_bounds__(128) void s4_conv_wmma(const float* __restrict__ Kk,
                                                    const float* __restrict__ XT,
                                                    float* __restrict__ YT) {
  __shared__ __align__(16) float Klds[Lc];        // K[h, 0..t0+63]
  __shared__ __align__(16) float Xlds[8 * 516];   // 512-col X panel, stride 516
  __shared__ float Ylds[64 * 9];
  const int h = blockIdx.x;
  const int t0 = blockIdx.y * 64;
  const int tid = threadIdx.x;
  const int w = tid >> 5, lane = tid & 31, n = lane & 15, hi = lane >> 4;

  const int kfloats = t0 + 64;                    // taps this block needs
  for (int i = tid; i < (kfloats >> 2); i += 128)
    async_b128_to_lds(&Klds[i << 2], Kk, (h * Lc + (i << 2)) * 4);

  v8f acc = {};
  const int tm = t0 + 16 * w + n;                 // this lane's output time row
  const int my_steps = (t0 + 16 * w + 16) >> 2;   // wave-uniform k-step count
  const int npanel = (t0 + 64 + 511) >> 9;
  const float* __restrict__ xh = XT + (size_t)h * Bc * Lc;

  for (int p = 0; p < npanel; ++p) {
    const int sbase = p << 9;
    for (int i = tid; i < 1024; i += 128) {       // stage 8 x 512 X panel
      int b = i >> 7, c4 = (i & 127) << 2;
      async_b128_to_lds(&Xlds[b * 516 + c4], xh, (b * Lc + sbase + c4) * 4);
    }
    async_wait0();
    __syncthreads();
    if (p + 1 < npanel)                           // warm L2 for next panel
      __builtin_prefetch(&xh[(tid >> 4) * Lc + sbase + 512 + (tid & 15) * 32], 0, 0);
    const int ks0 = sbase >> 2;
    int ks1 = ks0 + 128;
    if (ks1 > my_steps) ks1 = my_steps;
    for (int ks = ks0; ks < ks1; ++ks) {
      const int s = (ks << 2) + 2 * hi;
      const int d0 = tm - s, d1 = d0 - 1;
      v2f a;                                      // A = T[t, s] = K[t-s], causal
      a.x = (d0 >= 0) ? Klds[d0] : 0.0f;
      a.y = (d1 >= 0) ? Klds[d1] : 0.0f;
      const int sl = s - sbase;
      v2f b;                                      // B = X[s, n], n>=8 padded
      b.x = (n < 8) ? Xlds[n * 516 + sl] : 0.0f;
      b.y = (n < 8) ? Xlds[n * 516 + sl + 1] : 0.0f;
      acc = __builtin_amdgcn_wmma_f32_16x16x4_f32(
          false, a, false, b, (short)0, acc, false, false);
    }
    __syncthreads();
  }
  if (n < 8) {
#pragma unroll
    for (int r = 0; r < 8; ++r)
      Ylds[(16 * w + 8 * hi + r) * 9 + n] = acc[r];
  }
  __syncthreads();
  for (int i = tid; i < 512; i += 128) {          // coalesced tile writeout
    int b = i >> 6, tr = i & 63;
    YT[((size_t)h * Bc + b) * Lc + t0 + tr] = Ylds[tr * 9 + b];
  }
}

// ---------------------------------------------------------------- kernel 6 --
__global__ __launch_bounds__(256) void s4_transpose_out(const float* __restrict__ YT,
                                                        float* __restrict__ out) {
  __shared__ float tile[64][33];
  const int h0 = blockIdx.x * 32, t0 = blockIdx.y * 64, b = blockIdx.z;
  const int tid = threadIdx.x;
#pragma unroll
  for (int i = 0; i < 8; ++i) {
    int lin = tid + 256 * i;
    int hh = lin >> 6, ti = lin & 63;
    tile[ti][hh] = YT[((size_t)(h0 + hh) * Bc + b) * Lc + t0 + ti];
  }
  __syncthreads();
#pragma unroll
  for (int i = 0; i < 8; ++i) {
    int lin = tid + 256 * i;
    int ti = lin >> 5, hh = lin & 31;
    out[((size_t)(b * Lc + t0 + ti)) * Hc + h0 + hh] = tile[ti][hh];
  }
}

// ------------------------------------------------------------------ launch --
extern "C" void kernel_launch(void* const* d_in, const int* in_sizes, int n_in,
                              void* d_out, int out_size, void* d_ws, size_t ws_size,
                              hipStream_t stream) {
  const float* u   = (const float*)d_in[0];
  const float* Lr  = (const float*)d_in[1];
  const float* Li  = (const float*)d_in[2];
  const float* Pr  = (const float*)d_in[3];
  const float* Pi  = (const float*)d_in[4];
  const float* Brr = (const float*)d_in[5];
  const float* Bii = (const float*)d_in[6];
  const float* Cr  = (const float*)d_in[7];
  const float* Ci  = (const float*)d_in[8];
  const float* ldt = (const float*)d_in[9];
  const float* Dd  = (const float*)d_in[10];
  float* out = (float*)d_out;

  // Workspace layout (floats): AR | AIN | K | COS | SIN | XT | YT = 113.25 MB
  float* w   = (float*)d_ws;
  float* AR  = w;
  float* AIN = AR + HL;
  float* Kk  = AIN + HL;
  float* COS = Kk + HL;
  float* SIN = COS + LL;
  float* XT  = SIN + LL;
  float* YT  = XT + (size_t)BHL;

  s4_gen_tables<<<LL / 256, 256, 0, stream>>>(COS, SIN);
  s4_at_roots<<<Hc, 256, 0, stream>>>(Lr, Li, Pr, Pi, Brr, Bii, Cr, Ci, ldt, AR, AIN);
  s4_transpose_u<<<dim3(Hc / 32, Lc / 64, Bc), 256, 0, stream>>>(u, XT);
  s4_idft_wmma<<<dim3(Hc / 64, Lc / 64), 128, 0, stream>>>(AR, AIN, COS, SIN, Dd, Kk);
  s4_conv_wmma<<<dim3(Hc, Lc / 64), 128, 0, stream>>>(Kk, XT, YT);
  s4_transpose_out<<<dim3(Hc / 32, Lc / 64, Bc), 256, 0, stream>>>(YT, out);
}